// Net_66829691126192
// MI455X (gfx1250) — compile-verified
//
#include <hip/hip_runtime.h>

typedef __attribute__((ext_vector_type(2))) float v2f;
typedef __attribute__((ext_vector_type(8))) float v8f;

#define N_NODES 150000
#define N_EDGES 4800000
#define N_GRAPH 512
#define F_INF   4
#define HID     16

// ---------------------------------------------------------------------------
// init: deg = 1.0 (self-loop contribution), zero graph-pool accumulator
// ---------------------------------------------------------------------------
__global__ void k_init(float* __restrict__ deg, float* __restrict__ gpool) {
    int i = blockIdx.x * blockDim.x + threadIdx.x;
    if (i < N_NODES) deg[i] = 1.0f;
    if (i < N_GRAPH * HID) gpool[i] = 0.0f;
}

// degree accumulation over edge destinations (L2 atomics)
__global__ void k_deg(const int* __restrict__ dst, float* __restrict__ deg) {
    int e = blockIdx.x * blockDim.x + threadIdx.x;
    if (e < N_EDGES) atomicAdd(&deg[dst[e]], 1.0f);
}

// in-place deg -> deg^{-1/2}  (deg >= 1 always, self-loops included)
__global__ void k_dinv(float* __restrict__ deg) {
    int i = blockIdx.x * blockDim.x + threadIdx.x;
    if (i < N_NODES) deg[i] = rsqrtf(deg[i]);
}

// ---------------------------------------------------------------------------
// GEMM1: xw = x[N,4] @ W[4,16] via V_WMMA_F32_16X16X4_F32 (exact f32, K=4).
// One wave per 16-row tile. Fused epilogue: agg = xw * dinv^2 (self-loop msg).
// A layout (16x4 f32): lane m = l&15, VGPRs hold K = 2*(l>>4) + {0,1}.
// B layout (4x16 f32): lane n = l&15, VGPRs hold K = 2*(l>>4) + {0,1}.
// C/D layout: c[v] at row M = v + 8*(l>>4), col N = l&15.
// ---------------------------------------------------------------------------
__global__ void k_gemm4(const float* __restrict__ X, const float* __restrict__ W,
                        const float* __restrict__ dinv,
                        float* __restrict__ xw, float* __restrict__ agg) {
    int wave = (blockIdx.x * blockDim.x + threadIdx.x) >> 5;
    int lane = threadIdx.x & 31;
    const int tiles = N_NODES / 16;           // 9375, exact
    if (wave >= tiles) return;                // wave-uniform: EXEC all-ones below
    int row0 = wave * 16;
    int m  = lane & 15;
    int hi = lane >> 4;
    int k0 = hi * 2;
    int n  = lane & 15;

    v2f a = *(const v2f*)(X + (row0 + m) * F_INF + k0);
    v2f b; b.x = W[(k0 + 0) * HID + n]; b.y = W[(k0 + 1) * HID + n];
    v8f c = {};
    c = __builtin_amdgcn_wmma_f32_16x16x4_f32(false, a, false, b, (short)0, c,
                                              false, false);
#pragma unroll
    for (int v = 0; v < 8; ++v) {
        int r = row0 + v + 8 * hi;
        float d = dinv[r];
        xw[r * HID + n]  = c[v];
        agg[r * HID + n] = c[v] * d * d;      // self-loop message, seeds scatter
    }
}

// ---------------------------------------------------------------------------
// GEMM (K=16): xw = A[N,16] @ W[16,16], 4 chained WMMA f32 16x16x4 steps.
// Same fused self-loop epilogue.
// ---------------------------------------------------------------------------
__global__ void k_gemm16(const float* __restrict__ A, const float* __restrict__ W,
                         const float* __restrict__ dinv,
                         float* __restrict__ xw, float* __restrict__ agg) {
    int wave = (blockIdx.x * blockDim.x + threadIdx.x) >> 5;
    int lane = threadIdx.x & 31;
    const int tiles = N_NODES / 16;
    if (wave >= tiles) return;
    int row0 = wave * 16;
    int m  = lane & 15;
    int hi = lane >> 4;
    int k0 = hi * 2;
    int n  = lane & 15;

    v8f c = {};
#pragma unroll
    for (int s = 0; s < 4; ++s) {
        int kk = s * 4 + k0;
        v2f a = *(const v2f*)(A + (row0 + m) * HID + kk);
        v2f b; b.x = W[(kk + 0) * HID + n]; b.y = W[(kk + 1) * HID + n];
        c = __builtin_amdgcn_wmma_f32_16x16x4_f32(false, a, false, b, (short)0, c,
                                                  false, false);
    }
#pragma unroll
    for (int v = 0; v < 8; ++v) {
        int r = row0 + v + 8 * hi;
        float d = dinv[r];
        xw[r * HID + n]  = c[v];
        agg[r * HID + n] = c[v] * d * d;
    }
}

// ---------------------------------------------------------------------------
// Edge scatter: agg[dst] += xw[src] * dinv[src]*dinv[dst].
// One thread per edge: 4x b128 gathers (L2-resident), 16 f32 L2 atomics.
// ---------------------------------------------------------------------------
__global__ void k_scatter(const int* __restrict__ src, const int* __restrict__ dst,
                          const float* __restrict__ dinv,
                          const float* __restrict__ xw, float* __restrict__ agg) {
    int e = blockIdx.x * blockDim.x + threadIdx.x;
    if (e >= N_EDGES) return;
    int s = src[e], d = dst[e];
    float nrm = dinv[s] * dinv[d];
    const float4* p = (const float4*)(xw + (long)s * HID);
    float* q = agg + (long)d * HID;
#pragma unroll
    for (int j = 0; j < 4; ++j) {
        float4 v = p[j];
        atomicAdd(q + 4 * j + 0, v.x * nrm);
        atomicAdd(q + 4 * j + 1, v.y * nrm);
        atomicAdd(q + 4 * j + 2, v.z * nrm);
        atomicAdd(q + 4 * j + 3, v.w * nrm);
    }
}

// h = relu(agg + bias)
__global__ void k_bias_relu(const float* __restrict__ agg,
                            const float* __restrict__ bias,
                            float* __restrict__ h) {
    int i = blockIdx.x * blockDim.x + threadIdx.x;
    if (i < N_NODES * HID) {
        float v = agg[i] + bias[i & (HID - 1)];
        h[i] = fmaxf(v, 0.0f);
    }
}

// fused bias + relu + global_add_pool: gpool[batch[node]] += relu(agg + b2)
__global__ void k_pool(const float* __restrict__ agg, const float* __restrict__ b2,
                       const int* __restrict__ batch, float* __restrict__ gpool) {
    int i = blockIdx.x * blockDim.x + threadIdx.x;
    if (i >= N_NODES * HID) return;
    int node = i >> 4, col = i & 15;
    float v = fmaxf(agg[i] + b2[col], 0.0f);
    atomicAdd(&gpool[batch[node] * HID + col], v);
}

// ---------------------------------------------------------------------------
// Final MLP: out = relu(g @ Wf1 + bf1) @ Wf2 + bf2.  One wave per 16 graphs.
// WMMA for g@Wf1 (4x K=4 steps); second GEMM (16x1) done as a per-lane
// multiply by Wf2[n] + 16-lane shfl_xor reduction (xor masks 1,2,4,8 stay
// within each 16-lane half, matching the C-layout row split).
// ---------------------------------------------------------------------------
__global__ void k_mlp(const float* __restrict__ g, const float* __restrict__ Wf1,
                      const float* __restrict__ bf1, const float* __restrict__ Wf2,
                      const float* __restrict__ bf2, float* __restrict__ out) {
    int wave = (blockIdx.x * blockDim.x + threadIdx.x) >> 5;
    int lane = threadIdx.x & 31;
    const int tiles = N_GRAPH / 16;           // 32, exact
    if (wave >= tiles) return;
    int row0 = wave * 16;
    int m  = lane & 15;
    int hi = lane >> 4;
    int k0 = hi * 2;
    int n  = lane & 15;

    v8f c = {};
#pragma unroll
    for (int s = 0; s < 4; ++s) {
        int kk = s * 4 + k0;
        v2f a = *(const v2f*)(g + (row0 + m) * HID + kk);
        v2f b; b.x = Wf1[(kk + 0) * HID + n]; b.y = Wf1[(kk + 1) * HID + n];
        c = __builtin_amdgcn_wmma_f32_16x16x4_f32(false, a, false, b, (short)0, c,
                                                  false, false);
    }
    float w2 = Wf2[n];
    float bb = bf2[0];
    float b1n = bf1[n];
#pragma unroll
    for (int v = 0; v < 8; ++v) {
        float p = fmaxf(c[v] + b1n, 0.0f) * w2;
        p += __shfl_xor(p, 1, 32);
        p += __shfl_xor(p, 2, 32);
        p += __shfl_xor(p, 4, 32);
        p += __shfl_xor(p, 8, 32);
        if ((lane & 15) == 0) out[row0 + v + 8 * hi] = p + bb;
    }
}

// ---------------------------------------------------------------------------
extern "C" void kernel_launch(void* const* d_in, const int* in_sizes, int n_in,
                              void* d_out, int out_size, void* d_ws, size_t ws_size,
                              hipStream_t stream) {
    const float* x   = (const float*)d_in[0];
    const float* W1  = (const float*)d_in[1];
    const float* b1  = (const float*)d_in[2];
    const float* W2  = (const float*)d_in[3];
    const float* b2  = (const float*)d_in[4];
    const float* Wf1 = (const float*)d_in[5];
    const float* bf1 = (const float*)d_in[6];
    const float* Wf2 = (const float*)d_in[7];
    const float* bf2 = (const float*)d_in[8];
    const int*   ei  = (const int*)d_in[9];     // [2, E] flat
    const int*   bat = (const int*)d_in[10];
    const int* src = ei;
    const int* dst = ei + N_EDGES;
    float* out = (float*)d_out;

    // workspace partition (all 16B aligned): ~29.4 MB
    float* deg = (float*)d_ws;                  // N      (becomes dinv in-place)
    float* xw  = deg + N_NODES;                 // N*16   (reused layer 2)
    float* agg = xw  + (size_t)N_NODES * HID;   // N*16   (reused layer 2)
    float* h   = agg + (size_t)N_NODES * HID;   // N*16
    float* gp  = h   + (size_t)N_NODES * HID;   // G*16

    const int BLK = 256;
    int gN   = (N_NODES + BLK - 1) / BLK;
    int gE   = (N_EDGES + BLK - 1) / BLK;
    int gNH  = (N_NODES * HID + BLK - 1) / BLK;
    int gT   = ((N_NODES / 16) * 32 + BLK - 1) / BLK;   // 1 wave / 16-row tile
    int gM   = ((N_GRAPH / 16) * 32 + BLK - 1) / BLK;

    k_init     <<<gN,  BLK, 0, stream>>>(deg, gp);
    k_deg      <<<gE,  BLK, 0, stream>>>(dst, deg);
    k_dinv     <<<gN,  BLK, 0, stream>>>(deg);                    // deg -> dinv
    k_gemm4    <<<gT,  BLK, 0, stream>>>(x, W1, deg, xw, agg);    // layer 1 GEMM + self-loop seed
    k_scatter  <<<gE,  BLK, 0, stream>>>(src, dst, deg, xw, agg);
    k_bias_relu<<<gNH, BLK, 0, stream>>>(agg, b1, h);
    k_gemm16   <<<gT,  BLK, 0, stream>>>(h, W2, deg, xw, agg);    // layer 2 (buffers reused)
    k_scatter  <<<gE,  BLK, 0, stream>>>(src, dst, deg, xw, agg);
    k_pool     <<<gNH, BLK, 0, stream>>>(agg, b2, bat, gp);
    k_mlp      <<<gM,  BLK, 0, stream>>>(gp, Wf1, bf1, Wf2, bf2, out);
}